// STG_89721866813661
// MI455X (gfx1250) — compile-verified
//
#include <hip/hip_runtime.h>
#include <cstdint>
#include <cstddef>

// ---------------------------------------------------------------------------
// Problem constants (from reference): B=4, C_IN=HID=1536, OUT=512, N=64*64
// ---------------------------------------------------------------------------
#define BATCH     4
#define K_DIM     1536
#define N_NODES   4096
#define BM        128      // output-feature tile
#define BN        128      // node tile
#define BK        32       // K slab per async DMA burst
#define LDS_STRIDE 136     // 128 + 8 pad: 2*136 mod 64 == 16 -> conflict-free halves

typedef float v2f __attribute__((ext_vector_type(2)));
typedef float v8f __attribute__((ext_vector_type(8)));
typedef int   v4i __attribute__((ext_vector_type(4)));

// gfx1250 async global->LDS builtins (probe-confirmed present; fallback kept).
#if defined(__has_builtin)
#  if __has_builtin(__builtin_amdgcn_global_load_async_to_lds_b128) && \
      __has_builtin(__builtin_amdgcn_s_wait_asynccnt)
#    define HAVE_ASYNC_LDS 1
#  endif
#endif
#ifndef HAVE_ASYNC_LDS
#  define HAVE_ASYNC_LDS 0
#endif

#if HAVE_ASYNC_LDS
typedef __attribute__((address_space(1))) v4i* gv4i_ptr;
typedef __attribute__((address_space(3))) v4i* lv4i_ptr;
#endif

// ---------------------------------------------------------------------------
// Cooperative async fill of one BKxBM A-slab and one BKxBN B-slab into LDS.
// 256 threads x 4 float4 slots per matrix = 32 rows x 128 floats each.
// ---------------------------------------------------------------------------
__device__ __forceinline__ void fill_slab(
    const float* __restrict__ gA, const float* __restrict__ gB,
    float* dA, float* dB, int kb, int Fout, int tid)
{
  #pragma unroll
  for (int i = 0; i < 4; ++i) {
    int s  = tid + i * 256;
    int r  = s >> 5;             // 0..31  (k row)
    int c4 = (s & 31) << 2;      // 0..124 (col, float4 granular)
    const float* srcA = gA + (size_t)(kb + r) * Fout    + c4;
    const float* srcB = gB + (size_t)(kb + r) * N_NODES + c4;
    int d = r * LDS_STRIDE + c4;
#if HAVE_ASYNC_LDS
    __builtin_amdgcn_global_load_async_to_lds_b128(
        (gv4i_ptr)srcA, (lv4i_ptr)&dA[d], 0, 0);
    __builtin_amdgcn_global_load_async_to_lds_b128(
        (gv4i_ptr)srcB, (lv4i_ptr)&dB[d], 0, 0);
#else
    float4 va = *(const float4*)srcA;
    float4 vb = *(const float4*)srcB;
    *(float4*)&dA[d] = va;
    *(float4*)&dB[d] = vb;
#endif
  }
}

// ---------------------------------------------------------------------------
// Load one f32 WMMA operand fragment (16x4 A-tile or 4x16 B-tile; identical
// lane addressing): lanes 0-15 read k..k+1, lanes 16-31 read k+2..k+3.
// ---------------------------------------------------------------------------
__device__ __forceinline__ v2f load_frag(const float* s, int k, int col0,
                                         int laneLo, int laneHi) {
  int idx = (k + laneHi * 2) * LDS_STRIDE + col0 + laneLo;
  v2f f;
  f.x = s[idx];
  f.y = s[idx + LDS_STRIDE];
  return f;
}

// ---------------------------------------------------------------------------
// GEMM: out[b][j][n] = sum_c W[c][j] * in[b][c][n]  (+ bias[j] if non-null)
// A(16x4) = W^T tile, B(4x16) = activation tile, D(16x16) = (j, n) tile.
// Double-buffered: async DMA of slab kb+BK overlaps WMMA compute on slab kb;
// a single barrier per slab both publishes the filled buffer and retires
// reads of the buffer about to be refilled.
// Grid: (N_NODES/BN, Fout/BM, BATCH); 256 threads = 8 waves in 2(M)x4(N).
// ---------------------------------------------------------------------------
__global__ __launch_bounds__(256) void wmma_gemm_f32(
    const float* __restrict__ in,    // [B][K_DIM][N_NODES]
    const float* __restrict__ W,     // [K_DIM][Fout]
    const float* __restrict__ bias,  // [Fout] or nullptr
    float*       __restrict__ out,   // [B][Fout][N_NODES]
    int Fout)
{
  __shared__ float sA[2][BK * LDS_STRIDE];
  __shared__ float sB[2][BK * LDS_STRIDE];

  const int tid    = threadIdx.x;
  const int lane   = tid & 31;
  const int laneLo = lane & 15;
  const int laneHi = lane >> 4;
  const int wave   = tid >> 5;
  const int wm     = (wave >> 2) * 64;  // 0 / 64   : feature offset in tile
  const int wn     = (wave & 3) * 32;   // 0..96    : node offset in tile

  const int nb = blockIdx.x * BN;
  const int jb = blockIdx.y * BM;
  const int b  = blockIdx.z;

  const float* gA = W + (size_t)jb;                            // + c*Fout + j
  const float* gB = in + (size_t)b * K_DIM * N_NODES + nb;     // + c*N + n

  v8f acc[4][2] = {};

  // prologue: start DMA of first slab
  fill_slab(gA, gB, sA[0], sB[0], 0, Fout, tid);

  int p = 0;
  for (int kb = 0; kb < K_DIM; kb += BK, p ^= 1) {
#if HAVE_ASYNC_LDS
    __builtin_amdgcn_s_wait_asynccnt(0);
#endif
    __syncthreads();   // slab p complete everywhere; slab p^1 fully consumed

    if (kb + BK < K_DIM)
      fill_slab(gA, gB, sA[p ^ 1], sB[p ^ 1], kb + BK, Fout, tid);

    // --- compute: BK/4 = 8 k-steps, 8 WMMA tiles per wave per step ---------
    const float* cA = sA[p];
    const float* cB = sB[p];
    #pragma unroll
    for (int ks = 0; ks < BK; ks += 4) {
      v2f af[4], bf[2];
      #pragma unroll
      for (int mi = 0; mi < 4; ++mi)
        af[mi] = load_frag(cA, ks, wm + mi * 16, laneLo, laneHi);
      #pragma unroll
      for (int ni = 0; ni < 2; ++ni)
        bf[ni] = load_frag(cB, ks, wn + ni * 16, laneLo, laneHi);

      #pragma unroll
      for (int mi = 0; mi < 4; ++mi)
        #pragma unroll
        for (int ni = 0; ni < 2; ++ni)
          acc[mi][ni] = __builtin_amdgcn_wmma_f32_16x16x4_f32(
              false, af[mi], false, bf[ni],
              (short)0, acc[mi][ni], false, false);
    }
  }

  // --- epilogue: D tile rows are contiguous in n -> coalesced stores -------
  #pragma unroll
  for (int mi = 0; mi < 4; ++mi) {
    int jl = wm + mi * 16 + laneHi * 8;     // + v per element
    #pragma unroll
    for (int ni = 0; ni < 2; ++ni) {
      int nl = wn + ni * 16 + laneLo;
      size_t base = ((size_t)b * Fout + (size_t)(jb + jl)) * N_NODES + nb + nl;
      #pragma unroll
      for (int v = 0; v < 8; ++v) {
        float val = acc[mi][ni][v];
        if (bias) val += bias[jb + jl + v];
        out[base + (size_t)v * N_NODES] = val;
      }
    }
  }
}

// ---------------------------------------------------------------------------
// Degree normalization for the 4-connected 64x64 grid (+ self loop).
// ---------------------------------------------------------------------------
__global__ void norm_kernel(float* __restrict__ isd, float* __restrict__ invdeg) {
  int n = blockIdx.x * 256 + threadIdx.x;
  if (n >= N_NODES) return;
  int col = n & 63, row = n >> 6;
  float deg = 1.f + (col > 0) + (col < 63) + (row > 0) + (row < 63);
  isd[n]    = rsqrtf(deg);
  invdeg[n] = 1.f / deg;
}

// ---------------------------------------------------------------------------
// GCN aggregation = 5-point stencil over node dim with per-node norms.
// mode 1: out = relu(agg + bias) + extra   (extra = x identity)
// mode 2: out = (agg + bias) * extra       (extra = origin)
// ---------------------------------------------------------------------------
__global__ __launch_bounds__(256) void stencil_kernel(
    const float* __restrict__ h,       // [B][F][N_NODES]
    const float* __restrict__ isd,
    const float* __restrict__ invdeg,
    const float* __restrict__ bias,    // [F]
    const float* __restrict__ extra,   // [B][F][N_NODES]
    float*       __restrict__ out,
    int F, int total, int mode)
{
  int idx = blockIdx.x * 256 + threadIdx.x;
  if (idx >= total) return;
  int n   = idx & (N_NODES - 1);
  int f   = (idx >> 12) % F;
  int col = n & 63, row = n >> 6;

  float s   = isd[n];
  float val = h[idx] * invdeg[n];
  if (col > 0)  val += h[idx - 1]  * isd[n - 1]  * s;
  if (col < 63) val += h[idx + 1]  * isd[n + 1]  * s;
  if (row > 0)  val += h[idx - 64] * isd[n - 64] * s;
  if (row < 63) val += h[idx + 64] * isd[n + 64] * s;
  val += bias[f];

  out[idx] = (mode == 1) ? (fmaxf(val, 0.f) + extra[idx])
                         : (val * extra[idx]);
}

// ---------------------------------------------------------------------------
extern "C" void kernel_launch(void* const* d_in, const int* in_sizes, int n_in,
                              void* d_out, int out_size, void* d_ws, size_t ws_size,
                              hipStream_t stream) {
  (void)in_sizes; (void)n_in; (void)out_size; (void)ws_size;

  const float* x  = (const float*)d_in[0];   // [4][1536][4096]
  const float* W1 = (const float*)d_in[1];   // [1536][1536]
  const float* b1 = (const float*)d_in[2];
  const float* W2 = (const float*)d_in[3];   // [1536][512]
  const float* b2 = (const float*)d_in[4];
  const float* Wl = (const float*)d_in[5];   // [1536][512]
  const float* bl = (const float*)d_in[6];
  // d_in[7]=src, d_in[8]=dst: 4-connected grid, handled analytically.

  const int HID = 1536, OUTF = 512;

  char*  ws  = (char*)d_ws;
  size_t off = 0;
  auto carve = [&](size_t bytes) -> float* {
    float* p = (float*)(ws + off);
    off += (bytes + 255) & ~(size_t)255;
    return p;
  };
  float* isd    = carve((size_t)N_NODES * 4);
  float* invdeg = carve((size_t)N_NODES * 4);
  float* origin = carve((size_t)BATCH * OUTF * N_NODES * 4);   //  32 MiB
  float* h1     = carve((size_t)BATCH * HID  * N_NODES * 4);   //  96 MiB
  float* a      = carve((size_t)BATCH * HID  * N_NODES * 4);   //  96 MiB
  float* h2     = h1;  // h1 dead after stencil-1; reuse for GCN2 lin output

  // 1) grid normalization factors
  norm_kernel<<<(N_NODES + 255) / 256, 256, 0, stream>>>(isd, invdeg);

  // 2) origin = xt @ Wl + bl
  wmma_gemm_f32<<<dim3(N_NODES / BN, OUTF / BM, BATCH), 256, 0, stream>>>(
      x, Wl, bl, origin, OUTF);

  // 3) h1 = xt @ W1  (GCN1 linear, bias applied after aggregation)
  wmma_gemm_f32<<<dim3(N_NODES / BN, HID / BM, BATCH), 256, 0, stream>>>(
      x, W1, nullptr, h1, HID);

  // 4) a = relu(agg(h1) + b1) + x
  {
    int total = BATCH * HID * N_NODES;
    stencil_kernel<<<(total + 255) / 256, 256, 0, stream>>>(
        h1, isd, invdeg, b1, x, a, HID, total, 1);
  }

  // 5) h2 = a @ W2  (GCN2 linear)
  wmma_gemm_f32<<<dim3(N_NODES / BN, OUTF / BM, BATCH), 256, 0, stream>>>(
      a, W2, nullptr, h2, OUTF);

  // 6) out = (agg(h2) + b2) * origin
  {
    int total = BATCH * OUTF * N_NODES;
    stencil_kernel<<<(total + 255) / 256, 256, 0, stream>>>(
        h2, isd, invdeg, b2, origin, (float*)d_out, OUTF, total, 2);
  }
}